// ConvCaps_56977036149365
// MI455X (gfx1250) — compile-verified
//
#include <hip/hip_runtime.h>
#include <math.h>

typedef float v2f __attribute__((ext_vector_type(2)));
typedef float v4f __attribute__((ext_vector_type(4)));
typedef float v8f __attribute__((ext_vector_type(8)));

#define NB    8      // batch
#define NI    288    // B*K*K input-capsule x kernel positions
#define NJ    1152   // C*w*w output columns (c,xy)
#define NXY   36     // w*w
#define NROW  128    // C*4  (c, mr) GEMM rows
#define NCOL  1152   // NB*36*4 (b, xy, mc) GEMM cols
#define LOG_SQRT_2PI 0.9189385332046727f

// ---------------------------------------------------------------------------
// Kernel 0: activation patches a_in[b][i][xy] (independent of c)
// ---------------------------------------------------------------------------
__global__ void __launch_bounds__(256)
ain_kernel(const float* __restrict__ act, float* __restrict__ act_s) {
    int tid = blockIdx.x * 256 + threadIdx.x;
    if (tid >= NB * NI * NXY) return;
    int b  = tid / (NI * NXY);
    int r  = tid % (NI * NXY);
    int i  = r / NXY, xy = r % NXY;
    int bi = i / 9, kk = i % 9, k1 = kk / 3, k2 = kk % 3;
    int x  = xy / 6, y = xy % 6;
    act_s[tid] = act[((b * 32 + bi) * 14 + (2 * x + k1)) * 14 + (2 * y + k2)];
}

// ---------------------------------------------------------------------------
// Kernel 1: votes via V_WMMA_F32_16X16X4_F32.
// Per block i=(bi,k1,k2): D[128x1152] = A[128x4] (W slice) x B[4x1152] (poses).
// A fragment layout (16x4 f32): lanes 0-15 hold K=0,1 in .x/.y; lanes 16-31 K=2,3.
// B fragment (4x16): VGPR0 lanes0-15=K0 / lanes16-31=K2; VGPR1 = K1 / K3.
// D layout: VGPR v -> M = v + 8*(lane>=16), N = lane&15.
// ---------------------------------------------------------------------------
__global__ void __launch_bounds__(128)
votes_wmma(const float* __restrict__ poses, const float* __restrict__ W,
           float* __restrict__ V2) {
    const int i    = blockIdx.x;          // 0..287
    const int bi   = i / 9;
    const int kk   = i % 9;
    const int k1   = kk / 3, k2 = kk % 3;
    const int lane = threadIdx.x & 31;
    const int wave = threadIdx.x >> 5;    // 4 waves, 18 column tiles each
    const int lhalf = lane >> 4;          // 0 or 1
    const int l15   = lane & 15;

    // Preload all 8 A fragments for this W slice; reuse across all 72 col tiles.
    const float* Wslice = W + (size_t)i * 512;   // [c][mr][n] = 32*4*4
    v2f afrag[8];
#pragma unroll
    for (int rt = 0; rt < 8; ++rt) {
        int row = rt * 16 + l15;          // (c,mr)
        int kh  = lhalf * 2;
        afrag[rt].x = Wslice[row * 4 + kh];
        afrag[rt].y = Wslice[row * 4 + kh + 1];
    }

    for (int cti = 0; cti < 18; ++cti) {
        int ct  = wave * 18 + cti;
        int col = ct * 16 + l15;          // (bb, xy, mc)
        int bb  = col / 144;
        int rem = col % 144;
        int xy  = rem >> 2;
        int mc  = rem & 3;
        int x = xy / 6, y = xy % 6;
        int pr = 2 * x + k1, pc = 2 * y + k2;
        int n0 = lhalf * 2;
        // poses[bb, (n*4+mc)*32 + bi, pr, pc]
        const float* pbase = poses + (size_t)bb * 512 * 196 + (size_t)pr * 14 + pc;
        v2f bfrag;
        bfrag.x = pbase[((size_t)((n0    ) * 4 + mc) * 32 + bi) * 196];
        bfrag.y = pbase[((size_t)((n0 + 1) * 4 + mc) * 32 + bi) * 196];

#pragma unroll
        for (int rt = 0; rt < 8; ++rt) {
            v8f acc = {};
            acc = __builtin_amdgcn_wmma_f32_16x16x4_f32(
                false, afrag[rt], false, bfrag, (short)0, acc, false, false);
            float* vout = V2 + ((size_t)i * NROW + rt * 16) * NCOL + (size_t)ct * 16;
#pragma unroll
            for (int v = 0; v < 8; ++v) {
                int M = v + lhalf * 8;
                vout[(size_t)M * NCOL + l15] = acc[v];
            }
        }
    }
}

// ---------------------------------------------------------------------------
// Kernel 2: M-step. One thread per (b,j). Single pass moments over i=0..287:
// s0 = sum Rw, s1[m] = sum Rw*V, s2[m] = sum Rw*V^2 -> mu, sig2 = E[V^2]-mu^2.
// Per-lane V reads are 16B-aligned contiguous float4 -> coalesced b128 loads.
// Last iteration writes mu and a directly to d_out.
// ---------------------------------------------------------------------------
__global__ void __launch_bounds__(256)
mstep(const float* __restrict__ V2, const float* __restrict__ R,
      const float* __restrict__ act_s,
      const float* __restrict__ beta_v, const float* __restrict__ beta_a,
      const float* __restrict__ lambda_,
      float* __restrict__ mu, float* __restrict__ sig2,
      float* __restrict__ out_mu, float* __restrict__ out_a,
      int first_iter, int last_iter) {
    int tid = blockIdx.x * 256 + threadIdx.x;   // 0..9215 (exact)
    int b  = tid / NJ;
    int j  = tid % NJ;
    int c  = j / NXY;
    int xy = j % NXY;
    int colbase = b * 144 + xy * 4;             // multiple of 4 -> 16B aligned
    int rowbase = c * 4;
    const float rinit = 1.0f / (float)NJ;

    float s0 = 0.0f;
    float s1[16], s2[16];
#pragma unroll
    for (int m = 0; m < 16; ++m) { s1[m] = 0.0f; s2[m] = 0.0f; }

    for (int i = 0; i < NI; ++i) {
        float r  = first_iter ? rinit : R[((size_t)b * NI + i) * NJ + j];
        float rw = r * act_s[(b * NI + i) * NXY + xy];
        s0 += rw;
        const v4f* vb4 = (const v4f*)(V2 + ((size_t)i * NROW + rowbase) * NCOL + colbase);
        if (i + 1 < NI)
            __builtin_prefetch((const float*)vb4 + (size_t)NROW * NCOL, 0, 0);
#pragma unroll
        for (int mr = 0; mr < 4; ++mr) {
            v4f v = vb4[(size_t)mr * (NCOL / 4)];
#pragma unroll
            for (int mcq = 0; mcq < 4; ++mcq) {
                int   m  = mr * 4 + mcq;
                float vv = v[mcq];
                s1[m] = fmaf(rw, vv, s1[m]);
                s2[m] = fmaf(rw * vv, vv, s2[m]);
            }
        }
    }

    float inv0 = 1.0f / s0;
    float bv   = beta_v[0];
    float cost = 0.0f;
    v4f muo[4], sgo[4];
#pragma unroll
    for (int m = 0; m < 16; ++m) {
        float mum = s1[m] * inv0;
        float var = fmaxf(s2[m] * inv0 - mum * mum, 1e-30f);
        muo[m >> 2][m & 3] = mum;
        sgo[m >> 2][m & 3] = var;
        cost += (bv + __logf(var)) * s0;
    }
    v4f* mu4 = (v4f*)(mu   + (size_t)tid * 16);
    v4f* sg4 = (v4f*)(sig2 + (size_t)tid * 16);
#pragma unroll
    for (int q = 0; q < 4; ++q) { mu4[q] = muo[q]; sg4[q] = sgo[q]; }
    if (last_iter) {
        v4f* om4 = (v4f*)(out_mu + (size_t)tid * 16);
#pragma unroll
        for (int q = 0; q < 4; ++q) om4[q] = muo[q];
        float lam = lambda_[0], ba = beta_a[0];
        out_a[tid] = 1.0f / (1.0f + __expf(-lam * (ba - cost)));
    }
}

// ---------------------------------------------------------------------------
// Kernel 3: E-step. One workgroup per (b,i): ap[j] = a_in * sum_m N(V|mu,sig2),
// LDS-reduce the denominator over j, normalize R in place. V/mu/sig2 reads are
// vectorized b128.
// ---------------------------------------------------------------------------
__global__ void __launch_bounds__(256)
estep(const float* __restrict__ V2, const float* __restrict__ mu,
      const float* __restrict__ sig2, const float* __restrict__ act_s,
      float* __restrict__ R) {
    __shared__ float sdata[256];
    int b = blockIdx.x / NI;
    int i = blockIdx.x % NI;
    int t = threadIdx.x;

    float lsum = 0.0f;
    for (int j = t; j < NJ; j += 256) {
        int c  = j / NXY, xy = j % NXY;
        int colbase = b * 144 + xy * 4;
        int rowbase = c * 4;
        const v4f* vb4  = (const v4f*)(V2 + ((size_t)i * NROW + rowbase) * NCOL + colbase);
        const v4f* mub4 = (const v4f*)(mu   + ((size_t)b * NJ + j) * 16);
        const v4f* sgb4 = (const v4f*)(sig2 + ((size_t)b * NJ + j) * 16);
        float psum = 0.0f;
#pragma unroll
        for (int mr = 0; mr < 4; ++mr) {
            v4f v  = vb4[(size_t)mr * (NCOL / 4)];
            v4f mm = mub4[mr];
            v4f sg = sgb4[mr];
#pragma unroll
            for (int mcq = 0; mcq < 4; ++mcq) {
                float d  = v[mcq] - mm[mcq];
                float s  = sg[mcq];
                float lp = -0.5f * d * d / s - 0.5f * __logf(s) - LOG_SQRT_2PI;
                psum += __expf(lp);
            }
        }
        float ap = act_s[(b * NI + i) * NXY + xy] * psum;
        R[((size_t)b * NI + i) * NJ + j] = ap;
        lsum += ap;
    }
    sdata[t] = lsum;
    __syncthreads();
    for (int s = 128; s > 0; s >>= 1) {
        if (t < s) sdata[t] += sdata[t + s];
        __syncthreads();
    }
    float inv = 1.0f / sdata[0];
    for (int j = t; j < NJ; j += 256) {
        R[((size_t)b * NI + i) * NJ + j] *= inv;
    }
}

// ---------------------------------------------------------------------------
extern "C" void kernel_launch(void* const* d_in, const int* in_sizes, int n_in,
                              void* d_out, int out_size, void* d_ws, size_t ws_size,
                              hipStream_t stream) {
    const float* poses = (const float*)d_in[0];
    const float* act   = (const float*)d_in[1];
    const float* lam   = (const float*)d_in[2];
    const float* Wt    = (const float*)d_in[3];
    const float* bv    = (const float*)d_in[4];
    const float* ba    = (const float*)d_in[5];

    float* out = (float*)d_out;
    float* ws  = (float*)d_ws;

    float* V2     = ws;                                     // 288*128*1152
    float* R      = V2   + (size_t)NI * NROW * NCOL;        // 8*288*1152
    float* acts   = R    + (size_t)NB * NI * NJ;            // 8*288*36
    float* muW    = acts + (size_t)NB * NI * NXY;           // 8*1152*16
    float* sig2W  = muW  + (size_t)NB * NJ * 16;            // 8*1152*16
    float* out_mu = out;                                    // (8,32,6,6,16)
    float* out_a  = out + (size_t)NB * NJ * 16;             // (8,32,6,6)

    ain_kernel<<<(NB * NI * NXY + 255) / 256, 256, 0, stream>>>(act, acts);
    votes_wmma<<<NI, 128, 0, stream>>>(poses, Wt, V2);

    for (int it = 0; it < 3; ++it) {
        int first = (it == 0), last = (it == 2);
        mstep<<<(NB * NJ) / 256, 256, 0, stream>>>(
            V2, R, acts, bv, ba, lam, muW, sig2W, out_mu, out_a, first, last);
        if (!last)
            estep<<<NB * NI, 256, 0, stream>>>(V2, muW, sig2W, acts, R);
    }
}